// Decoder_17695265259993
// MI455X (gfx1250) — compile-verified
//
#include <hip/hip_runtime.h>
#include <hip/hip_bf16.h>

// ---------------------------------------------------------------------------
// Decoder: x[b] = mask_b * (query(16x8) @ z_b(8x8) @ map_w.T(8x128) + map_b)
// Memory-bound (512MB f32 output @ 23.3 TB/s ~ 23us). f32 WMMA 16x16x4 for
// the math (full precision, zero conversion cost); output staged per-wave in
// LDS so every global store is a contiguous 512B global_store_b128 burst.
// ---------------------------------------------------------------------------

typedef __attribute__((ext_vector_type(2))) float v2f;
typedef __attribute__((ext_vector_type(8))) float v8f;

#define NB      65536
#define NDIM    128
#define NWD     8
#define NP      16
#define ZSTRIDE 65          // 64 features + n
#define SSTR    132         // staged-row stride (pad: rows p, p+8 differ by 32 banks)
#define XELEMS  ((size_t)NB * NP * NDIM)
#define MELEMS  ((size_t)NB * NP)

// ---------------- kernel 0: query = pos(16x128) @ Wq_w.T(128x8) -------------
__global__ void query_kernel(const float* __restrict__ Wq, float* __restrict__ qout) {
    int t = threadIdx.x;            // 128 threads: one (p,k) each
    if (t >= NP * NWD) return;
    int p = t >> 3, k = t & 7;
    float acc = 0.f;
    const float neglog = -logf(10000.0f) / (float)NDIM;
    #pragma unroll 4
    for (int d = 0; d < NDIM; ++d) {
        int e = d & ~1;                         // even pair index
        float invf = __expf(neglog * (float)e);
        float ang  = (float)p * invf;
        float pe   = (d & 1) ? __cosf(ang) : __sinf(ang);
        acc += pe * Wq[k * NDIM + d];
    }
    qout[p * NWD + k] = acc;
}

// ---------------- main kernel: one wave handles a pair of batch rows --------
__global__ __launch_bounds__(256) void decoder_kernel(
    const float* __restrict__ z,      // [B,65]
    const float* __restrict__ map_w,  // [128,8] row-major; W[k][c] = map_w[c*8+k]
    const float* __restrict__ map_b,  // [128]
    const float* __restrict__ q,      // [16,8] query, from workspace
    float* __restrict__ out,          // x region: [B,16,128] f32
    int nPairs)
{
    __shared__ float tile[8][16 * 17];          // per-wave transpose tile
    __shared__ float stage[8][NP * SSTR];       // per-wave output stage (8.25KB)

    const int lane  = threadIdx.x & 31;
    const int wib   = threadIdx.x >> 5;         // wave in block (8 waves)
    const int gwave = blockIdx.x * 8 + wib;
    const int nWav  = gridDim.x * 8;
    float* T = tile[wib];
    float* S = stage[wib];

    const int col = lane & 15;                  // N index within 16-tile
    const int hi  = lane >> 4;                  // lane-half
    const int kb  = hi * 2;                     // K base for this lane-half

    // ---- preload W fragments (B-layout for 4x16 tiles) + bias, once per wave
    v2f wf0[8], wf1[8];
    float bv[8];
    #pragma unroll
    for (int j = 0; j < 8; ++j) {
        int c = j * 16 + col;
        wf0[j].x = map_w[c * 8 + kb];           // K = kb
        wf0[j].y = map_w[c * 8 + kb + 1];       // K = kb+1
        wf1[j].x = map_w[c * 8 + kb + 4];       // K = kb+4
        wf1[j].y = map_w[c * 8 + kb + 5];       // K = kb+5
        bv[j]    = map_b[j * 16 + col];
    }
    // ---- query A-fragments (A-layout for 16x4): lane holds row=col, K pair
    v2f qa0, qa1;
    qa0.x = q[col * 8 + kb];     qa0.y = q[col * 8 + kb + 1];
    qa1.x = q[col * 8 + kb + 4]; qa1.y = q[col * 8 + kb + 5];

    for (int pair = gwave; pair < nPairs; pair += nWav) {
        const int b0 = pair * 2, b1 = b0 + 1;

        if (pair + nWav < nPairs)               // global_prefetch_b8 next z rows
            __builtin_prefetch(&z[(size_t)(pair + nWav) * 2 * ZSTRIDE], 0, 1);

        // ---- gather z pair as B-matrix 8x16: cols 0-7 = b0, 8-15 = b1
        const int    zb   = (col < 8) ? b0 : b1;
        const size_t base = (size_t)zb * ZSTRIDE + (col & 7);   // + k*8
        v2f zf0, zf1;
        zf0.x = z[base + (size_t)(kb)     * 8];
        zf0.y = z[base + (size_t)(kb + 1) * 8];
        zf1.x = z[base + (size_t)(kb + 4) * 8];
        zf1.y = z[base + (size_t)(kb + 5) * 8];
        const float n0 = z[(size_t)b0 * ZSTRIDE + 64];
        const float n1 = z[(size_t)b1 * ZSTRIDE + 64];

        // ---- step 1: D2(16x16) = query(16x8) @ [z_b0 | z_b1](8x16)
        v8f d8 = {};
        d8 = __builtin_amdgcn_wmma_f32_16x16x4_f32(false, qa0, false, zf0,
                                                   (short)0, d8, false, false);
        d8 = __builtin_amdgcn_wmma_f32_16x16x4_f32(false, qa1, false, zf1,
                                                   (short)0, d8, false, false);

        // ---- C-layout -> A-layout transpose through per-wave LDS tile
        #pragma unroll
        for (int r = 0; r < 8; ++r)
            T[(r + 8 * hi) * 17 + col] = d8[r];
        asm volatile("s_wait_dscnt 0" ::: "memory");   // wave-private: no barrier

        #pragma unroll
        for (int bb = 0; bb < 2; ++bb) {
            const int   b  = bb ? b1 : b0;
            const float nb = bb ? n1 : n0;
            const int   cb = bb * 8;
            v2f a0, a1;                               // A-frags of D_b (16x8)
            a0.x = T[col * 17 + cb + kb];     a0.y = T[col * 17 + cb + kb + 1];
            a1.x = T[col * 17 + cb + kb + 4]; a1.y = T[col * 17 + cb + kb + 5];

            float mf[8];                              // row mask p < n_b
            #pragma unroll
            for (int r = 0; r < 8; ++r)
                mf[r] = ((float)(r + 8 * hi) < nb) ? 1.0f : 0.0f;

            // ---- step 2: X_b = D_b @ W + bias, masked, staged into LDS
            #pragma unroll
            for (int j = 0; j < 8; ++j) {
                v8f acc;                              // seed accumulator = bias
                #pragma unroll
                for (int r = 0; r < 8; ++r) acc[r] = bv[j];
                acc = __builtin_amdgcn_wmma_f32_16x16x4_f32(false, a0, false, wf0[j],
                                                            (short)0, acc, false, false);
                acc = __builtin_amdgcn_wmma_f32_16x16x4_f32(false, a1, false, wf1[j],
                                                            (short)0, acc, false, false);
                #pragma unroll
                for (int r = 0; r < 8; ++r)
                    S[(r + 8 * hi) * SSTR + j * 16 + col] = acc[r] * mf[r];
            }
            asm volatile("s_wait_dscnt 0" ::: "memory");

            // ---- stream out: one contiguous 512B global_store_b128 per row
            const size_t rowbase = (size_t)b * NP * NDIM;
            #pragma unroll
            for (int p = 0; p < NP; ++p) {
                float4 v = *(const float4*)&S[p * SSTR + lane * 4];
                *(float4*)&out[rowbase + (size_t)p * NDIM + lane * 4] = v;
            }
        }
    }
}

// ---------------- tail: mask + batch output regions -------------------------
__global__ void aux_kernel(const float* __restrict__ z, float* __restrict__ out) {
    int idx = blockIdx.x * blockDim.x + threadIdx.x;
    if (idx >= (int)MELEMS) return;
    int b = idx >> 4, p = idx & 15;
    float n = z[(size_t)b * ZSTRIDE + 64];
    out[XELEMS + idx]          = ((float)p < n) ? 1.0f : 0.0f;   // mask
    out[XELEMS + MELEMS + idx] = (float)b;                       // batch
}

extern "C" void kernel_launch(void* const* d_in, const int* in_sizes, int n_in,
                              void* d_out, int out_size, void* d_ws, size_t ws_size,
                              hipStream_t stream) {
    const float* z     = (const float*)d_in[0];   // [B,65]
    const float* Wq_w  = (const float*)d_in[1];   // [8,128]
    const float* map_w = (const float*)d_in[2];   // [128,8]
    const float* map_b = (const float*)d_in[3];   // [128]
    float* out = (float*)d_out;
    float* q   = (float*)d_ws;                    // 16x8 query scratch

    query_kernel<<<1, 128, 0, stream>>>(Wq_w, q);

    const int nPairs = NB / 2;                    // 32768 pairs, 1 per wave-iter
    decoder_kernel<<<1024, 256, 0, stream>>>(z, map_w, map_b, q, out, nPairs);

    aux_kernel<<<((int)MELEMS + 255) / 256, 256, 0, stream>>>(z, out);
}